// RotatedROIPool_68951404970591
// MI455X (gfx1250) — compile-verified
//
#include <hip/hip_runtime.h>
#include <hip/hip_bf16.h>

typedef __attribute__((ext_vector_type(2))) float v2f;
typedef __attribute__((ext_vector_type(8))) float v8f;

#define S_    64
#define C_    64
#define HF    128
#define WF    128
#define HWSZ  (HF * WF)
#define EXPANDF   1.25f
#define DEG2RADF  0.017453292519943295f

__device__ __forceinline__ int iclamp(int v, int lo, int hi) {
    return v < lo ? lo : (v > hi ? hi : v);
}

__global__ __launch_bounds__(256)
void RotatedROIPool_kernel(const float* __restrict__ feat,
                           const float* __restrict__ obb,
                           const int*   __restrict__ bix,
                           float*       __restrict__ out)
{
    // blockIdx.x in [0, 2N): n = bid>>1, (bid&1) selects tiles 0-7 vs 8-15.
    const int n    = blockIdx.x >> 1;
    const int tid  = threadIdx.x;
    const int wave = tid >> 5;
    const int lane = tid & 31;
    const int t    = ((blockIdx.x & 1) << 3) | wave;   // tile id 0..15
    const int ti   = t >> 2;                           // tile row 0..3
    const int tj   = t & 3;                            // tile col 0..3
    const int lm   = lane & 15;
    const int hv   = lane >> 4;                        // half of wave

    // ---- uniform ROI parameters (replicates obb_to_affine) ----
    const float scale = 1.0f / 8.0f;                   // 1/FEAT_DOWN
    const float cx = obb[n * 5 + 0] * scale;
    const float cy = obb[n * 5 + 1] * scale;
    const float ws = obb[n * 5 + 2] * scale * EXPANDF;
    const float hs = obb[n * 5 + 3] * scale * EXPANDF;
    const float th = obb[n * 5 + 4] * DEG2RADF;
    const float c_ = cosf(th);
    const float s_ = sinf(th);
    const float sx = ws * (1.0f / S_);
    const float sy = hs * (1.0f / S_);
    const float m00 = c_ * sx;
    const float m01 = -s_ * sy;
    const float m02 = cx;
    const float m10 = EXPANDF * sx;    // NOTE: reference uses EXPAND*sx here
    const float m11 = c_ * sy;
    const float m12 = cy;
    const int   b   = bix[n];

    // ---- affine grid for this 16x16 tile via V_WMMA_F32_16X16X4_F32 ----
    // G[i][j] = base_i*m01 + (m00*base_j + m02)   (and gy analogously)
    // A (16x4): A[i,0]=base_i, A[i,1]=1, A[i,2..3]=0.
    //   Layout: lanes 0-15 hold K=0,1 in VGPR0,1; lanes 16-31 hold K=2,3 -> zero.
    // B (4x16): row0 = m01 (const), row1 = m00*base_j + m02. Rows 2,3 don't
    //   matter (multiplied by zero A columns), so replicate on both halves.
    const float base_j = (2.0f * (float)(tj * 16 + lm) + 1.0f) * (1.0f / S_) - 1.0f;
    const float base_i = (2.0f * (float)(ti * 16 + lm) + 1.0f) * (1.0f / S_) - 1.0f;

    v2f Amat; Amat.x = hv ? 0.0f : base_i; Amat.y = hv ? 0.0f : 1.0f;
    v2f Bxm;  Bxm.x  = m01;  Bxm.y = m00 * base_j + m02;
    v2f Bym;  Bym.x  = m11;  Bym.y = m10 * base_j + m12;
    v8f zero = {};

    v8f gx = __builtin_amdgcn_wmma_f32_16x16x4_f32(
        false, Amat, false, Bxm, (short)0, zero, false, false);
    v8f gy = __builtin_amdgcn_wmma_f32_16x16x4_f32(
        false, Amat, false, Bym, (short)0, zero, false, false);

    // ---- per-point bilinear weights & clamped offsets (channel-invariant) ----
    // D layout: VGPR p = row ti*16 + p (lanes 0-15) or ti*16+8+p (lanes 16-31),
    // column j = tj*16 + (lane&15).
    const int j = tj * 16 + lm;
    int   off[8][4];
    float w[8][4];
    int   outOff[8];

#pragma unroll
    for (int p = 0; p < 8; ++p) {
        const float ix = ((gx[p] + 1.0f) * (float)WF - 1.0f) * 0.5f;
        const float iy = ((gy[p] + 1.0f) * (float)HF - 1.0f) * 0.5f;
        const float x0f = floorf(ix);
        const float y0f = floorf(iy);
        const float wx1 = ix - x0f;
        const float wy1 = iy - y0f;
        const float wx0 = 1.0f - wx1;
        const float wy0 = 1.0f - wy1;
        const int x0 = (int)x0f, y0 = (int)y0f;
        const int x1 = x0 + 1,  y1 = y0 + 1;

        const float vx0 = (x0 >= 0 && x0 < WF) ? 1.0f : 0.0f;
        const float vx1 = (x1 >= 0 && x1 < WF) ? 1.0f : 0.0f;
        const float vy0 = (y0 >= 0 && y0 < HF) ? 1.0f : 0.0f;
        const float vy1 = (y1 >= 0 && y1 < HF) ? 1.0f : 0.0f;

        const int x0c = iclamp(x0, 0, WF - 1);
        const int x1c = iclamp(x1, 0, WF - 1);
        const int y0c = iclamp(y0, 0, HF - 1);
        const int y1c = iclamp(y1, 0, HF - 1);

        w[p][0] = wy0 * wx0 * vy0 * vx0;
        w[p][1] = wy0 * wx1 * vy0 * vx1;
        w[p][2] = wy1 * wx0 * vy1 * vx0;
        w[p][3] = wy1 * wx1 * vy1 * vx1;

        off[p][0] = y0c * WF + x0c;
        off[p][1] = y0c * WF + x1c;
        off[p][2] = y1c * WF + x0c;
        off[p][3] = y1c * WF + x1c;

        const int ip = ti * 16 + (hv ? 8 : 0) + p;
        outOff[p] = ip * S_ + j;
    }

    // ---- channel loop: 4 gathers (L1/L2-resident feat) + 1 NT store ----
    const float* fb = feat + (size_t)b * C_ * HWSZ;
    float*       ob = out + (size_t)n * C_ * (S_ * S_);

    for (int c = 0; c < C_; ++c) {
        const float* fc = fb + c * HWSZ;
        float*       oc = ob + c * (S_ * S_);
        if (c + 8 < C_) {
            __builtin_prefetch(fc + 8 * HWSZ + off[0][0], 0, 0);
        }
#pragma unroll
        for (int p = 0; p < 8; ++p) {
            const float v = w[p][0] * fc[off[p][0]]
                          + w[p][1] * fc[off[p][1]]
                          + w[p][2] * fc[off[p][2]]
                          + w[p][3] * fc[off[p][3]];
            __builtin_nontemporal_store(v, oc + outOff[p]);
        }
    }
}

extern "C" void kernel_launch(void* const* d_in, const int* in_sizes, int n_in,
                              void* d_out, int out_size, void* d_ws, size_t ws_size,
                              hipStream_t stream) {
    const float* feat = (const float*)d_in[0];
    const float* obb  = (const float*)d_in[1];
    const int*   bix  = (const int*)d_in[2];
    float*       out  = (float*)d_out;

    const int N = in_sizes[2];            // number of ROIs (bix count)
    dim3 grid(2 * N);                     // 2 blocks per ROI (8 tiles each)
    dim3 block(256);                      // 8 wave32s, one 16x16 tile per wave
    hipLaunchKernelGGL(RotatedROIPool_kernel, grid, block, 0, stream,
                       feat, obb, bix, out);
}